// SingleHeadAttention_6889127543296
// MI455X (gfx1250) — compile-verified
//
#include <hip/hip_runtime.h>
#include <hip/hip_bf16.h>

// ---------------------------------------------------------------------------
// Single-head causal attention, fp32, for gfx1250 (MI455X).
//   B=4, S=4096, E=A=1024.
//   Kernel 1: Q/K/V projections via V_WMMA_F32_16X16X4_F32 (fp32 tensor op),
//             LDS tiles staged with GLOBAL_LOAD_ASYNC_TO_LDS_B128 (ASYNCcnt).
//   Kernel 2: fused flash-attention (QK^T, online softmax, PV), all matmuls
//             through v_wmma_f32_16x16x4_f32, wave32 layouts per CDNA5 ISA.
// Workspace: needs 3 * 16384 * 1024 * 4 B = 192 MB in d_ws for Q/K/V.
// ---------------------------------------------------------------------------

typedef float v2f __attribute__((ext_vector_type(2)));
typedef float v8f __attribute__((ext_vector_type(8)));

#define WMMA_F32(a, b, c) \
  __builtin_amdgcn_wmma_f32_16x16x4_f32(false, (a), false, (b), (short)0, (c), false, false)

// ---- CDNA5 async global->LDS copies (ASYNCcnt), with fallback --------------
#if __has_builtin(__builtin_amdgcn_global_load_async_to_lds_b32) && \
    __has_builtin(__builtin_amdgcn_global_load_async_to_lds_b128)
#define HAVE_ASYNC_LDS 1
#else
#define HAVE_ASYNC_LDS 0
#endif

#if HAVE_ASYNC_LDS
// Builtin signatures (from hipcc diagnostics):
//   b32 : (as(1) int*,                      as(3) int*,                      imm, imm)
//   b128: (as(1) vector_size(16) int*,      as(3) vector_size(16) int*,      imm, imm)
typedef int v4si __attribute__((vector_size(16)));
typedef __attribute__((address_space(1))) int*  gi_ptr;
typedef __attribute__((address_space(3))) int*  li_ptr;
typedef __attribute__((address_space(1))) v4si* gv4_ptr;
typedef __attribute__((address_space(3))) v4si* lv4_ptr;

__device__ __forceinline__ void async_g2l_b32(const void* g, void* l) {
  __builtin_amdgcn_global_load_async_to_lds_b32((gi_ptr)g, (li_ptr)l, 0, 0);
}
__device__ __forceinline__ void async_g2l_b128(const void* g, void* l) {
  __builtin_amdgcn_global_load_async_to_lds_b128((gv4_ptr)g, (lv4_ptr)l, 0, 0);
}
__device__ __forceinline__ void wait_async0() {
#if __has_builtin(__builtin_amdgcn_s_wait_asynccnt)
  __builtin_amdgcn_s_wait_asynccnt(0);
#else
  asm volatile("s_wait_asynccnt 0" ::: "memory");
#endif
}
#endif

static constexpr int Bq    = 16;    // query rows per block (attention)
static constexpr int Bk    = 128;   // key block size
static constexpr int Adim  = 1024;  // head dim
static constexpr int Sdim  = 4096;
static constexpr int Batch = 4;

// ---------------------------------------------------------------------------
// Projection GEMM: O[M,N] = X[M,K] * W[K,N], M=16384, K=N=1024.
// Block: 256 threads (8 waves), tile 128(M) x 128(N); wave tile 32x64
// (2x4 WMMA 16x16 accumulators -> 8 wmma per k-step for 6 fragment loads).
// blockIdx.z selects which projection (0:Q, 1:K, 2:V).
// ---------------------------------------------------------------------------
__global__ __launch_bounds__(256) void proj_gemm_kernel(
    const float* __restrict__ X,
    const float* __restrict__ Wq, const float* __restrict__ Wk,
    const float* __restrict__ Wv,
    float* __restrict__ Qo, float* __restrict__ Ko, float* __restrict__ Vo) {
  const float* W;
  float* O;
  if (blockIdx.z == 0)      { W = Wq; O = Qo; }
  else if (blockIdx.z == 1) { W = Wk; O = Ko; }
  else                      { W = Wv; O = Vo; }

  const int m0 = blockIdx.x * 128;
  const int n0 = blockIdx.y * 128;

  __shared__ float As[128 * 20];   // 128 rows x 16 k, pad->20 (80B rows, 16B aligned)
  __shared__ float Bs[16 * 128];   // 16 k x 128 n

  const int t    = threadIdx.x;
  const int lane = t & 31;
  const int w    = t >> 5;        // wave 0..7
  const int hf   = lane >> 4;     // half-wave select
  const int l16  = lane & 15;
  const int wm   = w >> 1;        // 0..3 : M sub-block (32 rows)
  const int wn   = w & 1;         // 0..1 : N sub-block (64 cols)

  const v8f vzero = {0.f, 0.f, 0.f, 0.f, 0.f, 0.f, 0.f, 0.f};
  v8f acc[2][4];
#pragma unroll
  for (int mi = 0; mi < 2; ++mi)
#pragma unroll
    for (int ni = 0; ni < 4; ++ni) acc[mi][ni] = vzero;

  for (int k0 = 0; k0 < 1024; k0 += 16) {
    __syncthreads();  // previous tile fully consumed
#if HAVE_ASYNC_LDS
    // A tile 128x16: 512 16B groups, 2 per thread (async, ASYNCcnt)
#pragma unroll
    for (int i = 0; i < 2; ++i) {
      int g = t + i * 256;
      int r = g >> 2, c4 = (g & 3) * 4;
      async_g2l_b128(&X[(size_t)(m0 + r) * 1024 + (k0 + c4)], &As[r * 20 + c4]);
    }
    // B tile 16x128: 512 16B groups, 2 per thread
#pragma unroll
    for (int i = 0; i < 2; ++i) {
      int g = t + i * 256;
      int r = g >> 5, c4 = (g & 31) * 4;
      async_g2l_b128(&W[(size_t)(k0 + r) * 1024 + (n0 + c4)], &Bs[r * 128 + c4]);
    }
    wait_async0();
#else
#pragma unroll
    for (int i = 0; i < 8; ++i) {
      int lin = t + i * 256;
      int r = lin >> 4, c = lin & 15;
      As[r * 20 + c] = X[(size_t)(m0 + r) * 1024 + (k0 + c)];
    }
#pragma unroll
    for (int i = 0; i < 8; ++i) {
      int lin = t + i * 256;
      int r = lin >> 7, c = lin & 127;
      Bs[r * 128 + c] = W[(size_t)(k0 + r) * 1024 + (n0 + c)];
    }
#endif
    if (k0 + 16 < 1024)
      __builtin_prefetch(&X[(size_t)(m0 + (t >> 1)) * 1024 + (k0 + 16)], 0, 1);
    __syncthreads();

#pragma unroll
    for (int kk = 0; kk < 16; kk += 4) {
      // A fragments (16x4 fp32): lane = M row, half selects K-pair {0,1}/{2,3}
      v2f afr[2], bfr[4];
#pragma unroll
      for (int mi = 0; mi < 2; ++mi) {
        int ar = wm * 32 + mi * 16 + l16;
        afr[mi].x = As[ar * 20 + kk + 2 * hf];
        afr[mi].y = As[ar * 20 + kk + 2 * hf + 1];
      }
      // B fragments (4x16 fp32): VGPR v holds rows {v, v+2} split by half
#pragma unroll
      for (int ni = 0; ni < 4; ++ni) {
        int bc = wn * 64 + ni * 16 + l16;
        bfr[ni].x = Bs[(kk + 2 * hf) * 128 + bc];
        bfr[ni].y = Bs[(kk + 2 * hf + 1) * 128 + bc];
      }
#pragma unroll
      for (int mi = 0; mi < 2; ++mi)
#pragma unroll
        for (int ni = 0; ni < 4; ++ni)
          acc[mi][ni] = WMMA_F32(afr[mi], bfr[ni], acc[mi][ni]);
    }
  }

  // Store: C/D layout -> lane = N col, VGPR v = row v (+8 for upper half)
#pragma unroll
  for (int mi = 0; mi < 2; ++mi)
#pragma unroll
    for (int ni = 0; ni < 4; ++ni)
#pragma unroll
      for (int v = 0; v < 8; ++v) {
        int row = m0 + wm * 32 + mi * 16 + v + 8 * hf;
        int col = n0 + wn * 64 + ni * 16 + l16;
        O[(size_t)row * 1024 + col] = acc[mi][ni][v];
      }
}

// ---------------------------------------------------------------------------
// Fused causal flash attention.
// Block: 256 threads (8 waves), 16 query rows, full 1024-wide output.
// Wave w: computes score tile for keys [w*16, w*16+16) of each 128-key block,
// and owns output feature columns [w*128, (w+1)*128) (8 fp32 accumulators).
// Dynamic LDS layout (floats):
//   Qs [16][1028] (row 4112B, 16B-aligned for async b128; banks l16*4 mod 64)
//   Kt [64][129]  K chunk transposed (b32 async scatter)
//   Pt [16][129]  probabilities
//   pmax[8][16], psum[8][16], mrun/lrun/alph/mnew[16]
// ---------------------------------------------------------------------------
static constexpr int QS_STRIDE = 1028;
static constexpr int QS_OFF   = 0;                       // 16*1028 = 16448
static constexpr int KT_OFF   = 16448;                   // 64*129  = 8256
static constexpr int PT_OFF   = 24704;                   // 16*129  = 2064
static constexpr int PMAX_OFF = 26768;
static constexpr int PSUM_OFF = 26896;
static constexpr int MRUN_OFF = 27024;
static constexpr int LRUN_OFF = 27040;
static constexpr int ALPH_OFF = 27056;
static constexpr int MNEW_OFF = 27072;
static constexpr int SMEM_FLOATS = 27088;

__global__ __launch_bounds__(256) void attn_fused_kernel(
    const float* __restrict__ Q, const float* __restrict__ K,
    const float* __restrict__ V, float* __restrict__ out) {
  const int qb0 = blockIdx.x * Bq;
  const int b   = blockIdx.y;
  const float* Qb = Q + (size_t)b * Sdim * Adim;
  const float* Kb = K + (size_t)b * Sdim * Adim;
  const float* Vb = V + (size_t)b * Sdim * Adim;

  extern __shared__ float smem[];
  float* Qs   = smem + QS_OFF;
  float* Kt   = smem + KT_OFF;
  float* Pt   = smem + PT_OFF;
  float* pmax = smem + PMAX_OFF;
  float* psum = smem + PSUM_OFF;
  float* mrun = smem + MRUN_OFF;
  float* lrun = smem + LRUN_OFF;
  float* alph = smem + ALPH_OFF;
  float* mnew = smem + MNEW_OFF;

  const int t    = threadIdx.x;
  const int lane = t & 31;
  const int w    = t >> 5;
  const int hf   = lane >> 4;
  const int l16  = lane & 15;

  // Stage Q rows (unscaled; 1/sqrt(A) applied to raw scores later)
#if HAVE_ASYNC_LDS
#pragma unroll 4
  for (int i = 0; i < 16; ++i) {
    int g = t + i * 256;
    int r = g >> 8, c4 = (g & 255) * 4;
    async_g2l_b128(&Qb[(size_t)(qb0 + r) * Adim + c4], &Qs[r * QS_STRIDE + c4]);
  }
  wait_async0();
#else
#pragma unroll 4
  for (int i = 0; i < 64; ++i) {
    int lin = t + i * 256;
    int r = lin >> 10, c = lin & 1023;
    Qs[r * QS_STRIDE + c] = Qb[(size_t)(qb0 + r) * Adim + c];
  }
#endif
  if (t < 16) { mrun[t] = -1e30f; lrun[t] = 0.f; }
  __syncthreads();

  const v8f vzero = {0.f, 0.f, 0.f, 0.f, 0.f, 0.f, 0.f, 0.f};
  v8f oacc[8];
#pragma unroll
  for (int i = 0; i < 8; ++i) oacc[i] = vzero;

  const int nkb = (qb0 >> 7) + 1;  // causal: key blocks with keys <= qb0+15
  for (int kb = 0; kb < nkb; ++kb) {
    const int kbase = kb << 7;

    // ---- scores: S_raw = Q K^T, accumulated over A in 64-wide chunks -------
    v8f sc = vzero;
    for (int ac = 0; ac < Adim; ac += 64) {
      __syncthreads();  // previous Kt chunk fully consumed
      // Stage K[kbase..+128][ac..+64] transposed -> Kt[a][key]
#if HAVE_ASYNC_LDS
#pragma unroll 4
      for (int i = 0; i < 32; ++i) {
        int lin = t + i * 256;
        int key = lin >> 6, a = lin & 63;
        async_g2l_b32(&Kb[(size_t)(kbase + key) * Adim + (ac + a)],
                      &Kt[a * 129 + key]);
      }
      wait_async0();
#else
#pragma unroll 4
      for (int i = 0; i < 32; ++i) {
        int lin = t + i * 256;
        int key = lin >> 6, a = lin & 63;
        Kt[a * 129 + key] = Kb[(size_t)(kbase + key) * Adim + (ac + a)];
      }
#endif
      __syncthreads();
#pragma unroll
      for (int aa = 0; aa < 64; aa += 4) {
        v2f af, bf;
        af.x = Qs[l16 * QS_STRIDE + ac + aa + 2 * hf];
        af.y = Qs[l16 * QS_STRIDE + ac + aa + 2 * hf + 1];
        bf.x = Kt[(aa + 2 * hf) * 129 + w * 16 + l16];
        bf.y = Kt[(aa + 2 * hf + 1) * 129 + w * 16 + l16];
        sc = WMMA_F32(af, bf, sc);
      }
    }

    // ---- scale + causal mask + per-row max (reduce across 16-lane halves) --
    const float scale = 0.03125f;  // 1/sqrt(1024)
    float s[8];
    const int keyg = kbase + w * 16 + l16;  // this lane's key column
    const bool lastblk = (kb == nkb - 1);
#pragma unroll
    for (int v = 0; v < 8; ++v) {
      float x = sc[v] * scale;
      int qg = qb0 + v + 8 * hf;
      if (lastblk && keyg > qg) x = -1e30f;
      s[v] = x;
    }
#pragma unroll
    for (int v = 0; v < 8; ++v) {
      float m = s[v];
      for (int off = 1; off < 16; off <<= 1) m = fmaxf(m, __shfl_xor(m, off));
      if (l16 == 0) pmax[w * 16 + v + 8 * hf] = m;  // lanes 0 and 16 write
    }
    __syncthreads();

    // ---- online softmax state update (wave 0, one thread per row) ----------
    if (t < 16) {
      float mo = mrun[t];
      float mb = mo;
#pragma unroll
      for (int w2 = 0; w2 < 8; ++w2) mb = fmaxf(mb, pmax[w2 * 16 + t]);
      mnew[t] = mb;
      alph[t] = __expf(mo - mb);
      mrun[t] = mb;
    }
    __syncthreads();

    // ---- P = exp(S - mnew), partial row sums, stash P in LDS ---------------
#pragma unroll
    for (int v = 0; v < 8; ++v) {
      int row = v + 8 * hf;
      float pv = __expf(s[v] - mnew[row]);
      Pt[row * 129 + w * 16 + l16] = pv;
      float r = pv;
      for (int off = 1; off < 16; off <<= 1) r += __shfl_xor(r, off);
      if (l16 == 0) psum[w * 16 + row] = r;
    }
    __syncthreads();

    if (t < 16) {
      float l2 = lrun[t] * alph[t];
#pragma unroll
      for (int w2 = 0; w2 < 8; ++w2) l2 += psum[w2 * 16 + t];
      lrun[t] = l2;
    }

    // ---- rescale O accumulators by alpha ----------------------------------
    float al8[8];
#pragma unroll
    for (int v = 0; v < 8; ++v) al8[v] = alph[v + 8 * hf];
#pragma unroll
    for (int ni = 0; ni < 8; ++ni)
#pragma unroll
      for (int v = 0; v < 8; ++v) oacc[ni][v] *= al8[v];

    // ---- O += P * V  (V read coalesced from global; L2-resident) ----------
    for (int k0 = 0; k0 < Bk; k0 += 4) {
      v2f af;
      af.x = Pt[l16 * 129 + k0 + 2 * hf];
      af.y = Pt[l16 * 129 + k0 + 2 * hf + 1];
      const float* vrow0 = Vb + (size_t)(kbase + k0 + 2 * hf) * Adim;
      const float* vrow1 = vrow0 + Adim;
#pragma unroll
      for (int ni = 0; ni < 8; ++ni) {
        int n = w * 128 + ni * 16 + l16;
        v2f bf;
        bf.x = vrow0[n];
        bf.y = vrow1[n];
        oacc[ni] = WMMA_F32(af, bf, oacc[ni]);
      }
    }
    __syncthreads();  // Pt/alph/lrun stable before next key block
  }

  // ---- epilogue: normalize, round to 4 decimals (round-half-even), store ---
  float linv[8];
#pragma unroll
  for (int v = 0; v < 8; ++v) linv[v] = 1.0f / lrun[v + 8 * hf];
#pragma unroll
  for (int ni = 0; ni < 8; ++ni)
#pragma unroll
    for (int v = 0; v < 8; ++v) {
      float x = oacc[ni][v] * linv[v];
      x = rintf(x * 1.0e4f) * 1.0e-4f;
      size_t row = (size_t)b * Sdim + qb0 + v + 8 * hf;
      out[row * Adim + w * 128 + ni * 16 + l16] = x;
    }
}

// ---------------------------------------------------------------------------
extern "C" void kernel_launch(void* const* d_in, const int* in_sizes, int n_in,
                              void* d_out, int out_size, void* d_ws, size_t ws_size,
                              hipStream_t stream) {
  // setup_inputs order: embedded, Wk, Wq, Wv (all fp32)
  const float* X  = (const float*)d_in[0];
  const float* Wk = (const float*)d_in[1];
  const float* Wq = (const float*)d_in[2];
  const float* Wv = (const float*)d_in[3];
  float* out = (float*)d_out;

  const size_t MN = (size_t)Batch * Sdim * Adim;  // 16 Mi elements
  float* Qbuf = (float*)d_ws;       // needs 192 MB total in d_ws
  float* Kbuf = Qbuf + MN;
  float* Vbuf = Kbuf + MN;

  // Projections: grid (M/128, N/128, 3)
  proj_gemm_kernel<<<dim3(128, 8, 3), 256, 0, stream>>>(
      X, Wq, Wk, Wv, Qbuf, Kbuf, Vbuf);

  // Fused attention: grid (S/16, B), ~106 KB dynamic LDS per block
  const size_t smem_bytes = (size_t)SMEM_FLOATS * sizeof(float);
  (void)hipFuncSetAttribute((const void*)attn_fused_kernel,
                            hipFuncAttributeMaxDynamicSharedMemorySize,
                            (int)smem_bytes);
  attn_fused_kernel<<<dim3(Sdim / Bq, Batch), 256, smem_bytes, stream>>>(
      Qbuf, Kbuf, Vbuf, out);
}